// MaskConv2d_67791763800306
// MI455X (gfx1250) — compile-verified
//
#include <hip/hip_runtime.h>
#include <hip/hip_bf16.h>

// MI455X (gfx1250): fused masked-conv stack as implicit GEMM on
// V_WMMA_F32_16X16X4_F32 (wave32, fp32-precise matrix pipe).
//
// Roofline: 0.53 TFLOP fp32, ~0.7 GB HBM traffic (~30us at 23.3 TB/s);
// weights (<1MB) are L2-resident, so the binding resource is fp32 matrix
// compute -> inner loop is pure ds_load_b64 + v_wmma, constant DS offsets.
// TT=160 divides T=800 exactly: zero wasted time-tiles (TT=128 wasted ~12%).
//
// Layouts (per documented f32 WMMA striping):
//   A (16x4): lane<16 -> (M=lane, K=+0/1), lane>=16 -> (M=lane-16, K=+2/3)
//   B (4x16): lane<16 -> (K=+0/1, N=lane), lane>=16 -> (K=+2/3, N=lane-16)
//   C/D: vgpr r: lane<16 -> (M=r, N=lane), lane>=16 -> (M=r+8, N=lane-16)
//
// Weight slabs stored [co][kh][12] (kw 11 padded to 12, pad zeroed) so every
// K-chunk of 4 is (kh, kwc*4..kwc*4+3): A pair and B pair are consecutive
// LDS words at compile-time offsets from two per-lane base pointers.

typedef float v2f __attribute__((ext_vector_type(2)));
typedef float v8f __attribute__((ext_vector_type(8)));

#define TT       160          // output time positions per block (800 = 5*160)
#define NTHR     640          // 20 wave32s: (2 co-halves) x (10 time sub-tiles)
#define T_OUT    800
#define NB       32           // batch
// conv1 geometry
#define C1_KH    41
#define C1_KWP   12
#define C1_KROW  (C1_KH * C1_KWP)   // 492 floats per co (even -> 8B-aligned rows)
#define H_IN     161
#define T_IN     1600
#define H1       81
#define SX1_COLS 332          // 2*TT + 10 = 330, padded even
// conv2 geometry
#define C2_KH    21
#define C2_KWP   12
#define C2_KSLAB (C2_KH * C2_KWP)   // 252 floats per co per ci
#define H2       41
#define SX2_COLS 172          // TT + 10 = 170, padded even

__device__ __forceinline__ int len_after_conv1(int L) {
    // floor((L + 2*5 - (11-1) - 1)/2) + 1, clamped at 0  ==  L>0 ? (L-1)/2+1 : 0
    return (L > 0) ? (((L - 1) >> 1) + 1) : 0;
}

// ---------------------------------------------------------------- conv1 ----
__global__ __launch_bounds__(NTHR) void conv1_wmma(
    const float* __restrict__ x, const int* __restrict__ seq,
    const float* __restrict__ W1, const float* __restrict__ b1,
    float* __restrict__ out1)
{
    extern __shared__ float smem[];
    float* sW = smem;                      // [32][41][12]  (kw==11 zeroed)
    float* sX = smem + 32 * C1_KROW;       // [41][SX1_COLS] zero-padded input tile

    const int tid = threadIdx.x;
    const int t0  = blockIdx.x * TT;
    const int ho  = blockIdx.y;
    const int b   = blockIdx.z;

    for (int i = tid; i < 32 * C1_KROW; i += NTHR) {
        int co = i / C1_KROW, k = i - co * C1_KROW;
        int kh = k / C1_KWP,  kw = k - kh * C1_KWP;
        sW[i] = (kw < 11) ? W1[co * 451 + kh * 11 + kw] : 0.0f;
    }
    for (int i = tid; i < C1_KH * SX1_COLS; i += NTHR) {
        int r = i / SX1_COLS, c = i - r * SX1_COLS;
        int hi = 2 * ho - 20 + r;          // pad_h = 20
        int gt = 2 * t0 - 5 + c;           // pad_w = 5, stride_w = 2
        float v = 0.0f;
        if (hi >= 0 && hi < H_IN && gt >= 0 && gt < T_IN)
            v = x[(size_t)(b * H_IN + hi) * T_IN + gt];
        sX[i] = v;
    }
    __syncthreads();

    const int wave    = tid >> 5;
    const int lane    = tid & 31;
    const int lo      = lane & 15;
    const int hi16    = lane >> 4;             // 0: K+0/1 | M+0 ; 1: K+2/3 | M+8
    const int co_base = (wave & 1) * 16;
    const int tt      = (wave >> 1) * 16 + lo; // local time col (B / store N)

    // Per-lane base pointers: everything else is a constant DS offset.
    const float* aP = sW + (co_base + lo) * C1_KROW + 2 * hi16;
    const float* bP = sX + 2 * tt + 2 * hi16;

    v8f acc = {};
    for (int kh = 0; kh < C1_KH; ++kh) {
#pragma unroll
        for (int kwc = 0; kwc < 3; ++kwc) {
            v2f a  = *(const v2f*)(aP + kwc * 4);   // ds_load_b64
            v2f bb = *(const v2f*)(bP + kwc * 4);   // ds_load_b64 (2*tt even)
            acc = __builtin_amdgcn_wmma_f32_16x16x4_f32(false, a, false, bb,
                                                        (short)0, acc, false, false);
        }
        aP += C1_KWP;
        bP += SX1_COLS;
    }

    const int len1 = len_after_conv1(seq[b]);
    const int t    = t0 + tt;                   // always < 800 (exact tiling)
    const float mask = (t < len1) ? 1.0f : 0.0f;
    for (int r = 0; r < 8; ++r) {
        int co  = co_base + r + 8 * hi16;
        float v = acc[r] + b1[co];
        v = (v > 0.0f ? v : 0.0f) * mask;       // mask->relu->mask == relu-then-mask
        out1[((size_t)(b * 32 + co) * H1 + ho) * T_OUT + t] = v;
    }
}

// ---------------------------------------------------------------- conv2 ----
__global__ __launch_bounds__(NTHR) void conv2_wmma(
    const float* __restrict__ in1, const int* __restrict__ seq,
    const float* __restrict__ W2, const float* __restrict__ b2,
    float* __restrict__ out)
{
    extern __shared__ float smem[];
    float* sW = smem;                      // [32][21][12] per-ci slab (kw==11 zeroed)
    float* sX = smem + 32 * C2_KSLAB;      // [21][SX2_COLS] per-ci input rows

    const int tid = threadIdx.x;
    const int t0  = blockIdx.x * TT;
    const int ho  = blockIdx.y;
    const int b   = blockIdx.z;

    const int wave    = tid >> 5;
    const int lane    = tid & 31;
    const int lo      = lane & 15;
    const int hi16    = lane >> 4;
    const int co_base = (wave & 1) * 16;
    const int tt      = (wave >> 1) * 16 + lo;
    const float* sWm  = sW + (co_base + lo) * C2_KSLAB + 2 * hi16;
    const float* sXn  = sX + tt + 2 * hi16;

    v8f acc = {};
    for (int ci = 0; ci < 32; ++ci) {
        __syncthreads();                    // previous slab fully consumed
        for (int i = tid; i < 32 * C2_KSLAB; i += NTHR) {
            int co = i / C2_KSLAB, k = i - co * C2_KSLAB;
            int kh = k / C2_KWP,  kw = k - kh * C2_KWP;
            sW[i] = (kw < 11) ? W2[((size_t)(co * 32 + ci) * C2_KH + kh) * 11 + kw]
                              : 0.0f;
        }
        for (int i = tid; i < C2_KH * SX2_COLS; i += NTHR) {
            int r = i / SX2_COLS, c = i - r * SX2_COLS;
            int hi = 2 * ho - 10 + r;       // pad_h = 10
            int gt = t0 - 5 + c;            // pad_w = 5, stride_w = 1
            float v = 0.0f;
            if (hi >= 0 && hi < H1 && gt >= 0 && gt < T_OUT)
                v = in1[((size_t)(b * 32 + ci) * H1 + hi) * T_OUT + gt];
            sX[i] = v;                      // in1 already masked+relu'd
        }
        // Prefetch next ci's input slab (128B per participating thread) so the
        // next fill hits cache: emits global_prefetch_b8.
        if (ci + 1 < 32) {
            int pi = tid * 32;              // 128B stride across the slab
            if (pi < C2_KH * SX2_COLS) {
                int r = pi / SX2_COLS, c = pi - r * SX2_COLS;
                int hi = 2 * ho - 10 + r;
                int gt = t0 - 5 + c;
                if (hi >= 0 && hi < H1 && gt >= 0 && gt < T_OUT)
                    __builtin_prefetch(
                        &in1[((size_t)(b * 32 + ci + 1) * H1 + hi) * T_OUT + gt], 0, 3);
            }
        }
        __syncthreads();

        const float* aP = sWm;
        const float* bP = sXn;
        for (int kh = 0; kh < C2_KH; ++kh) {
#pragma unroll
            for (int kwc = 0; kwc < 3; ++kwc) {
                v2f a = *(const v2f*)(aP + kwc * 4);    // ds_load_b64
                v2f bb;                                  // tt parity varies: b32 pair
                bb.x = bP[kwc * 4];
                bb.y = bP[kwc * 4 + 1];
                acc = __builtin_amdgcn_wmma_f32_16x16x4_f32(false, a, false, bb,
                                                            (short)0, acc, false, false);
            }
            aP += C2_KWP;
            bP += SX2_COLS;
        }
    }

    const int len2 = len_after_conv1(seq[b]);   // conv2 leaves lengths unchanged
    const int t    = t0 + tt;                   // always < 800 (exact tiling)
    const float mask = (t < len2) ? 1.0f : 0.0f;
    for (int r = 0; r < 8; ++r) {
        int co  = co_base + r + 8 * hi16;
        float v = acc[r] + b2[co];
        v = (v > 0.0f ? v : 0.0f) * mask;
        out[((size_t)(b * 32 + co) * H2 + ho) * T_OUT + t] = v;
    }
}

// ------------------------------------------------------------- lens tail ---
__global__ void lens_kernel(const int* __restrict__ seq, float* __restrict__ tail)
{
    int i = threadIdx.x;
    if (i < NB) tail[i] = (float)len_after_conv1(seq[i]);
}

// ---------------------------------------------------------------- launch ---
extern "C" void kernel_launch(void* const* d_in, const int* in_sizes, int n_in,
                              void* d_out, int out_size, void* d_ws, size_t ws_size,
                              hipStream_t stream)
{
    const float* x   = (const float*)d_in[0];
    const int*   seq = (const int*)  d_in[1];
    const float* W1  = (const float*)d_in[2];
    const float* b1  = (const float*)d_in[3];
    const float* W2  = (const float*)d_in[4];
    const float* b2  = (const float*)d_in[5];
    float* out = (float*)d_out;
    float* ws1 = (float*)d_ws;   // masked+relu'd conv1 output: 32*32*81*800 f32 (~253 MiB)

    const dim3 blk(NTHR);
    const dim3 g1(T_OUT / TT, H1, NB);   // 5 x 81 x 32, exact tiling
    const size_t smem1 = (size_t)(32 * C1_KROW + C1_KH * SX1_COLS) * sizeof(float); // ~117 KB
    conv1_wmma<<<g1, blk, smem1, stream>>>(x, seq, W1, b1, ws1);

    const dim3 g2(T_OUT / TT, H2, NB);   // 5 x 41 x 32, exact tiling
    const size_t smem2 = (size_t)(32 * C2_KSLAB + C2_KH * SX2_COLS) * sizeof(float); // ~46 KB
    conv2_wmma<<<g2, blk, smem2, stream>>>(ws1, seq, W2, b2, out);

    lens_kernel<<<1, 32, 0, stream>>>(seq, out + (size_t)NB * 32 * H2 * T_OUT);
}